// OuterProductLayer_47820165874257
// MI455X (gfx1250) — compile-verified
//
#include <hip/hip_runtime.h>
#include <stdint.h>

#define FIELDS 32
#define EMB    64
#define PITCH  33                 // transposed LDS pitch (dwords): 33 coprime 64 -> conflict-free
#define TRI    2016               // 64*63/2
#define OUT_PER_B (FIELDS * TRI)  // 64512

// CDNA5 async global->LDS (ASYNCcnt) if the toolchain exposes the builtin.
#if __has_builtin(__builtin_amdgcn_global_load_async_to_lds_b32)
#define USE_ASYNC_LDS 1
#else
#define USE_ASYNC_LDS 0
#endif

#if defined(__AMDGCN__)
typedef __attribute__((address_space(1))) int* gptr_i32;
typedef __attribute__((address_space(3))) int* lptr_i32;
#endif

__device__ __forceinline__ int tri_off(int i) {  // rank of pair (i, i+1) in triu(64, k=1)
  return (i * (127 - i)) >> 1;
}

__global__ void __launch_bounds__(256)
outer_flat_triu_kernel(const float* __restrict__ in, float* __restrict__ out) {
  __shared__ float Xs[EMB * PITCH];  // Xs[e*33 + d] = x[b][d][e]  (transposed, padded)

  const int tid = threadIdx.x;
  const int b   = blockIdx.x;
  const float* __restrict__ xb = in + (size_t)b * (FIELDS * EMB);

  // ---------------- stage x[b] (32x64 f32) into LDS, transposed ----------------
  // global element l = d*64 + e  ->  Xs[e*33 + d]
#pragma unroll
  for (int s = 0; s < 8; ++s) {
    const int l = tid + 256 * s;  // 0..2047, lane-contiguous -> coalesced global reads
    const int d = l >> 6;
    const int e = l & 63;
    float* dst = &Xs[e * PITCH + d];
#if USE_ASYNC_LDS
    float* gsrc = const_cast<float*>(xb + l);
    __builtin_amdgcn_global_load_async_to_lds_b32(
        (gptr_i32)gsrc, (lptr_i32)dst, 0, 0);
#else
    *dst = xb[l];
#endif
  }
#if USE_ASYNC_LDS
  asm volatile("s_wait_asynccnt 0" ::: "memory");
#endif
  __syncthreads();

  // ---------------- compute ----------------
  // out[b][f][p(i,j)] = X[j&31][2f + (i>>5)] * X[j&31][2*(i&31) + (j>>5)],  i<j
  // wave k: jhi = k&1, j = 32*jhi + lane (so d = lane); i residue class w = k>>1 (mod 4)
  const int lane = tid & 31;
  const int wv   = __builtin_amdgcn_readfirstlane(tid >> 5);  // wave id 0..7 (scalar)
  const int jhi  = wv & 1;
  const int w    = wv >> 1;

  // B-side values, loaded once, reused for all 32 f:
  // bv[m] = X[d=lane][2*(w+4m) + jhi]
  float bv[8];
#pragma unroll
  for (int m = 0; m < 8; ++m)
    bv[m] = Xs[(2 * (w + 4 * m) + jhi) * PITCH + lane];

  // wave-uniform store-offset constants (kept scalar):
  //   low  slots: i = w+4m  (<32):  p = off(i) + j - i - 1 = soffL[m] + lane
  //   high slots: i = 32+w+4m    :  p = soffH[m] + lane   (jhi==1 waves only)
  int iL[8], soffL[8], soffH[8];
#pragma unroll
  for (int m = 0; m < 8; ++m) {
    const int i  = w + 4 * m;
    const int ih = i + 32;
    iL[m]    = i;
    soffL[m] = tri_off(i)  - i  - 1 + (jhi << 5);
    soffH[m] = tri_off(ih) - ih - 1 + 32;
  }

  float* __restrict__ ob = out + (size_t)b * OUT_PER_B;

  if (jhi == 0) {
    // j = lane in [0,32): only pairs with lane > i exist in this half
    for (int f = 0; f < 32; ++f) {
      const float a0 = Xs[(2 * f) * PITCH + lane];  // h = 2f (i<32 always here)
      float* of = ob + f * TRI;
#pragma unroll
      for (int m = 0; m < 8; ++m)
        if (lane > iL[m])
          __builtin_nontemporal_store(a0 * bv[m], of + soffL[m] + lane);
    }
  } else {
    // j = 32 + lane
    for (int f = 0; f < 32; ++f) {
      const float a0 = Xs[(2 * f)     * PITCH + lane];  // h = 2f   (i < 32)
      const float a1 = Xs[(2 * f + 1) * PITCH + lane];  // h = 2f+1 (i >= 32)
      float* of = ob + f * TRI;
#pragma unroll
      for (int m = 0; m < 8; ++m)  // i < 32: j = 32+lane > i always -> dense, unpredicated
        __builtin_nontemporal_store(a0 * bv[m], of + soffL[m] + lane);
#pragma unroll
      for (int m = 0; m < 8; ++m)  // i = 32+w+4m: need lane > w+4m
        if (lane > iL[m])
          __builtin_nontemporal_store(a1 * bv[m], of + soffH[m] + lane);
    }
  }
}

extern "C" void kernel_launch(void* const* d_in, const int* in_sizes, int n_in,
                              void* d_out, int out_size, void* d_ws, size_t ws_size,
                              hipStream_t stream) {
  const float* in = (const float*)d_in[0];
  float* out      = (float*)d_out;
  const int batch = in_sizes[0] / (FIELDS * EMB);  // 1024
  outer_flat_triu_kernel<<<batch, 256, 0, stream>>>(in, out);
}